// threshold_89223650607168
// MI455X (gfx1250) — compile-verified
//
#include <hip/hip_runtime.h>
#include <hip/hip_bf16.h>

// Canny non-maximum suppression, fp32, B=8 C=1 H=2048 W=2048 (shapes fixed by
// the reference -> baked in as constexpr so row strides fold to immediates).
//
// Roofline: 33.55M px * (8B read + 4B write) = ~402 MB compulsory HBM traffic
// -> ~17 us floor at 23.3 TB/s. Stencil reuse (3 rows of g) lives in L2
// (a row is 8 KB, L2 is 192 MB). Pure bandwidth kernel:
//   - b128 loads/stores (float4 per lane)
//   - 32-bit voffset indexing (SADDR base + VGPR offset, no 64-bit VALU math)
//   - branch-free direction logic: 4 shared v_cmp + cndmask-selected neighbor
//     pair, EXEC stays full (wave32, no divergence)
//   - branchless (s_cselect-clamped) global_prefetch_b8, 4 rows ahead into L2

namespace {

constexpr float kD225  = 0.39269908169872414f;  // pi/8
constexpr float kD675  = 1.1780972450961724f;   // 3pi/8
constexpr float kD1125 = 1.9634954084936207f;   // 5pi/8
constexpr float kD1575 = 2.7488935718910690f;   // 7pi/8

// One pixel. ta = |t|. Selects the neighbor pair along the quantized gradient
// direction with 3 cndmask levels, then 2 compares + 1 select for the result.
__device__ __forceinline__ float nms1(float gv, float ta,
                                      float l,  float r,
                                      float tp, float bt,
                                      float tl, float tr,
                                      float bl, float br) {
  const bool c225  = ta >= kD225;
  const bool c675  = ta >= kD675;
  const bool c1125 = ta >= kD1125;
  const bool c1575 = ta >= kD1575;
  const bool h     = (!c225) || c1575;     // [0,pi/8) U [7pi/8, pi]
  const bool d45   = c225 && !c675;        // [pi/8, 3pi/8)
  const bool v     = c675 && !c1125;       // [3pi/8, 5pi/8)
  // default (none of the above) = d135 in [5pi/8, 7pi/8) — categories are
  // disjoint and exhaustive for ta in [0, pi].
  float n1 = tl, n2 = br;                  // d135 pair
  n1 = v   ? tp : n1;  n2 = v   ? bt : n2;
  n1 = d45 ? tr : n1;  n2 = d45 ? bl : n2;
  n1 = h   ? l  : n1;  n2 = h   ? r  : n2;
  const bool keep = (gv >= n1) && (gv >= n2);
  return keep ? gv : 0.0f;
}

} // namespace

__global__ __launch_bounds__(256) void nms_kernel(
    const float* __restrict__ g,
    const float* __restrict__ t,
    float*       __restrict__ out) {
  constexpr unsigned H = 2048, W = 2048, NROWS = 8u * 2048u;  // B*H

  const unsigned row  = blockIdx.y;                 // 0 .. NROWS-1
  const unsigned irow = row & (H - 1u);             // row within image
  const unsigned c0   = ((blockIdx.x * blockDim.x) + threadIdx.x) << 2;

  // All offsets fit in 32 bits (max ~2^25 elements) -> SADDR + voffset mode.
  const unsigned base  = row * W;
  const unsigned tbase = (irow == 0u)      ? base : base - W;  // edge clamp
  const unsigned bbase = (irow == H - 1u)  ? base : base + W;  // edge clamp

  // Prefetch 4 rows ahead into L2, branchless: clamp via scalar cselect
  // (row is wave-uniform). At the tail this re-prefetches the current row —
  // a harmless L2 hit — instead of paying a scalar branch diamond.
  const unsigned pbase = (row + 4u < NROWS) ? base + 4u * W : base;
  __builtin_prefetch(g + pbase + c0, 0, 1);
  __builtin_prefetch(t + pbase + c0, 0, 1);

  // 128-bit vector loads; c0*4 bytes is 16B-aligned, rows are 8 KB apart.
  const float4 gc = *reinterpret_cast<const float4*>(g + base  + c0);  // center
  const float4 tv = *reinterpret_cast<const float4*>(t + base  + c0);  // theta
  const float4 gt = *reinterpret_cast<const float4*>(g + tbase + c0);  // top
  const float4 gb = *reinterpret_cast<const float4*>(g + bbase + c0);  // bottom

  // Horizontal halo (edge-clamped). L0/L2 hits — same lines as neighbors' b128s.
  const unsigned cl = (c0 == 0u)        ? 0u     : c0 - 1u;
  const unsigned cr = (c0 + 4u >= W)    ? W - 1u : c0 + 4u;
  const float hgl = g[base  + cl], hgr = g[base  + cr];
  const float htl = g[tbase + cl], htr = g[tbase + cr];
  const float hbl = g[bbase + cl], hbr = g[bbase + cr];

  float4 o;
  //            g      |t|          left  right  top   bot   t-left t-right b-left b-right
  o.x = nms1(gc.x, fabsf(tv.x), hgl,  gc.y, gt.x, gb.x, htl,   gt.y,  hbl,   gb.y);
  o.y = nms1(gc.y, fabsf(tv.y), gc.x, gc.z, gt.y, gb.y, gt.x,  gt.z,  gb.x,  gb.z);
  o.z = nms1(gc.z, fabsf(tv.z), gc.y, gc.w, gt.z, gb.z, gt.y,  gt.w,  gb.y,  gb.w);
  o.w = nms1(gc.w, fabsf(tv.w), gc.z, hgr,  gt.w, gb.w, gt.z,  htr,   gb.z,  hbr);

  *reinterpret_cast<float4*>(out + base + c0) = o;  // b128 store
}

extern "C" void kernel_launch(void* const* d_in, const int* in_sizes, int n_in,
                              void* d_out, int out_size, void* d_ws, size_t ws_size,
                              hipStream_t stream) {
  const float* g = (const float*)d_in[0];
  const float* t = (const float*)d_in[1];
  float* out     = (float*)d_out;

  constexpr int B = 8, H = 2048, W = 2048;
  constexpr int NROWS = B * H;                 // 16384
  constexpr int THREADS = 256;
  constexpr int XBLOCKS = W / (4 * THREADS);   // 2 blocks of float4 per row

  dim3 block(THREADS, 1, 1);
  dim3 grid(XBLOCKS, NROWS, 1);                // (2, 16384)
  nms_kernel<<<grid, block, 0, stream>>>(g, t, out);
}